// PRMS_21492016349469
// MI455X (gfx1250) — compile-verified
//
#include <hip/hip_runtime.h>
#include <hip/hip_bf16.h>
#include <math.h>

#define T_STEPS 730
#define NZF 1e-5f

// Hardware base-2 transcendentals (v_log_f32 / v_exp_f32).
#if defined(__has_builtin)
#if __has_builtin(__builtin_amdgcn_logf) && __has_builtin(__builtin_amdgcn_exp2f)
#define FAST_LOG2(x) __builtin_amdgcn_logf(x)
#define FAST_EXP2(x) __builtin_amdgcn_exp2f(x)
#endif
#endif
#ifndef FAST_LOG2
#define FAST_LOG2(x) log2f(x)
#define FAST_EXP2(x) exp2f(x)
#endif

// bounds for: tt ddf alpha beta stor retip fscn scx flz stot cgw resmax k1..k6
__constant__ static const float c_LB[18] = {
    -3.f, 0.f, 0.f, 0.f, 0.f, 0.f, 0.f, 0.f, 0.005f, 1.f, 0.f, 1.f,
    0.f, 1.f, 0.f, 0.f, 0.f, 0.f};
__constant__ static const float c_UB[18] = {
    5.f, 20.f, 1.f, 1.f, 5.f, 50.f, 1.f, 1.f, 0.995f, 2000.f, 20.f, 300.f,
    1.f, 5.f, 1.f, 1.f, 1.f, 1.f};

// One PRMS time step. Consumes forcing (P_, T_, E_), updates the 7 state
// registers, and produces q. Faithful transcription of the JAX step incl.
// every max(..., NZ) clamp.
#define PRMS_STEP(P_, T_, E_, Q_)                                              \
  do {                                                                         \
    const float cold = ((T_) <= tt) ? 1.0f : 0.0f;                             \
    const float flux_ps = (P_)*cold;                                           \
    const float flux_pr = (P_)-flux_ps; /* == P*(1-cold) exactly */            \
    snow = snow + flux_ps;                                                     \
    const float flux_m = fminf(fmaxf(ddf * ((T_)-tt), 0.0f), snow);            \
    snow = fmaxf(snow - flux_m, NZF);                                          \
    const float flux_pim = flux_pr * one_m_beta;                               \
    const float flux_psm = flux_pr * beta;                                     \
    const float flux_pby = flux_psm * one_m_alpha;                             \
    const float flux_pin = flux_psm * alpha;                                   \
    xin = xin + flux_pin;                                                      \
    const float flux_ptf = fmaxf(xin - stor, 0.0f);                            \
    xin = fmaxf(xin - flux_ptf, NZF);                                          \
    const float flux_ein = fminf((E_)*beta, xin);                              \
    xin = fmaxf(xin - flux_ein, NZF);                                          \
    const float flux_mim = flux_m * one_m_beta;                                \
    const float flux_msm = flux_m * beta;                                      \
    rstor = rstor + flux_mim + flux_pim;                                       \
    const float flux_sas = fmaxf(rstor - retip, 0.0f);                         \
    rstor = fmaxf(rstor - flux_sas, NZF);                                      \
    const float flux_eim = fminf(one_m_beta * (E_), rstor);                    \
    rstor = fmaxf(rstor - flux_eim, NZF);                                      \
    const float sro_ratio = fminf(fmaxf(scn + rechr * sro_c, 0.0f), 1.0f);     \
    const float inflow = flux_msm + flux_ptf + flux_pby;                       \
    const float flux_sro = sro_ratio * inflow;                                 \
    const float flux_inf = inflow - flux_sro;                                  \
    rechr = rechr + flux_inf;                                                  \
    const float flux_pc = fmaxf(rechr - remx, 0.0f);                           \
    rechr = rechr - flux_pc;                                                   \
    const float ep_rem = (E_)-flux_ein - flux_eim;                             \
    const float evap_max_a = fmaxf(rechr * inv_remx * ep_rem, 0.0f);           \
    const float flux_ea = fminf(evap_max_a, rechr);                            \
    rechr = fmaxf(rechr - flux_ea, NZF);                                       \
    smav = smav + flux_pc;                                                     \
    const float flux_excs = fmaxf(smav - smax, 0.0f);                          \
    smav = smav - flux_excs;                                                   \
    float transp =                                                             \
        (rechr < ep_rem) ? (smav * inv_smax * (ep_rem - flux_ea)) : 0.0f;      \
    transp = fmaxf(transp, 0.0f);                                              \
    smav = fmaxf(smav - transp, NZF);                                          \
    const float flux_sep = fminf(cgw, flux_excs);                              \
    const float flux_qres = fmaxf(flux_excs - flux_sep, 0.0f);                 \
    res = res + flux_qres;                                                     \
    /* res > 0 always => pow via hw log2/exp2 (v_log_f32 / v_exp_f32) */       \
    const float pw = FAST_EXP2(k2 * FAST_LOG2(res * inv_resmax));              \
    const float flux_gad = fminf(k1 * pw, res);                                \
    res = fmaxf(res - flux_gad, NZF);                                          \
    const float flux_ras = fminf(k3 * res + k4 * res * res, res);              \
    res = fmaxf(res - flux_ras, NZF);                                          \
    gw = gw + flux_gad + flux_sep;                                             \
    const float flux_bas = k5 * gw;                                            \
    gw = fmaxf(gw - flux_bas, NZF);                                            \
    const float flux_snk = k6 * gw;                                            \
    gw = fmaxf(gw - flux_snk, NZF);                                            \
    (Q_) = flux_sas + flux_sro + flux_bas + flux_ras;                          \
  } while (0)

__global__ __launch_bounds__(32) void prms_scan_kernel(
    const float* __restrict__ x,       // [T, G, 3]
    const float* __restrict__ params,  // [T, G, 18, 1]
    float* __restrict__ out,           // [T, G, 1]
    int G) {
  const int g = blockIdx.x * blockDim.x + threadIdx.x;
  if (g >= G) return;

  // ---- unbound static parameters (taken at t = T-1, nmul = 0) ----
  const float* __restrict__ pbase =
      params + (size_t)(T_STEPS - 1) * (size_t)G * 18 + (size_t)g * 18;
  float sp[18];
#pragma unroll
  for (int i = 0; i < 18; ++i) sp[i] = c_LB[i] + pbase[i] * (c_UB[i] - c_LB[i]);

  const float tt = sp[0], ddf = sp[1], alpha = sp[2], beta = sp[3];
  const float stor = sp[4], retip = sp[5], fscn = sp[6], scx = sp[7];
  const float flz = sp[8], stot = sp[9], cgw = sp[10], resmax = sp[11];
  const float k1 = sp[12], k2 = sp[13], k3 = sp[14], k4 = sp[15];
  const float k5 = sp[16], k6 = sp[17];

  const float scn = fscn * scx;
  const float remx = (1.0f - flz) * stot;  // > 0 given bounds
  const float smax = flz * stot;           // > 0 given bounds
  const float inv_remx = 1.0f / remx;
  const float inv_smax = 1.0f / smax;
  const float inv_resmax = 1.0f / resmax;
  const float one_m_beta = 1.0f - beta;
  const float one_m_alpha = 1.0f - alpha;
  const float sro_c = (scx - scn) * inv_remx;  // sro = clamp(scn + rechr*sro_c)

  // ---- state (registers) ----
  float snow = 0.001f, xin = 0.001f, rstor = 0.001f, rechr = 0.001f;
  float smav = 0.001f, res = 0.001f, gw = 0.001f;

  const size_t strideT = (size_t)G * 3;  // elements per time row of x
  const float* __restrict__ xg = x + (size_t)g * 3;

  // software pipeline: current forcing in registers; next row loaded during
  // compute via an incrementing pointer (no per-iteration index math)
  float P = xg[0], Tt = xg[1], Ep = xg[2];
  const float* __restrict__ xptr = xg + strideT;  // row t+1
  const size_t pfOff = 7 * strideT;               // xptr+pfOff == row t+8
  float* __restrict__ optr = out + (size_t)g;     // q[t, g]

  // Region A: t in [0, T-9) -- prefetch 8 rows ahead, no per-iter predicate
  int t = 0;
  for (; t < T_STEPS - 9; ++t) {
    __builtin_prefetch(xptr + pfOff, 0, 3);  // gfx1250 global_prefetch_b8
    const float Pn = xptr[0], Tn = xptr[1], En = xptr[2];  // global_load_b96
    xptr += strideT;

    float q;
    PRMS_STEP(P, Tt, Ep, q);
    __builtin_nontemporal_store(q, optr);  // coalesced, write-once stream
    optr += (size_t)G;

    P = Pn; Tt = Tn; Ep = En;  // rotate pipeline
  }

  // Region B: t in [T-9, T-1) -- everything already resident, no prefetch
  for (; t < T_STEPS - 1; ++t) {
    const float Pn = xptr[0], Tn = xptr[1], En = xptr[2];
    xptr += strideT;

    float q;
    PRMS_STEP(P, Tt, Ep, q);
    __builtin_nontemporal_store(q, optr);
    optr += (size_t)G;

    P = Pn; Tt = Tn; Ep = En;
  }

  // peeled final step (no next-row load)
  float q;
  PRMS_STEP(P, Tt, Ep, q);
  __builtin_nontemporal_store(q, optr);
}

extern "C" void kernel_launch(void* const* d_in, const int* in_sizes, int n_in,
                              void* d_out, int out_size, void* d_ws,
                              size_t ws_size, hipStream_t stream) {
  (void)n_in; (void)d_ws; (void)ws_size; (void)out_size;
  const float* x = (const float*)d_in[0];       // [730, 8192, 3] f32
  const float* params = (const float*)d_in[1];  // [730, 8192, 18, 1] f32
  float* out = (float*)d_out;                   // [730, 8192, 1] f32

  const int G = in_sizes[0] / (T_STEPS * 3);  // 8192
  const int block = 32;                       // single-wave workgroups
  const int grid = (G + block - 1) / block;   // 256 workgroups, max WGP spread
  prms_scan_kernel<<<grid, block, 0, stream>>>(x, params, out, G);
}